// erosion_and_dilation_74345883894463
// MI455X (gfx1250) — compile-verified
//
#include <hip/hip_runtime.h>
#include <stdint.h>

// 3x3 erosion (min) + dilation (max), NCHW (8,32,512,512) fp32, replicate border.
// Memory-bound: ~830 MiB traffic -> ~36us floor at 23.3 TB/s. Input staged to LDS
// via the gfx1250 async global->LDS path (global_load_async_to_lds_b128 /
// s_wait_asynccnt); outputs streamed with non-temporal b128 stores so the 512 MiB
// write stream does not evict the input-row reuse window in L2.

#define IMG_H 512
#define IMG_W 512
#define TILE_H 16
#define TILE_W 128
#define LDS_ROWS 18          // TILE_H + 2 halo rows
#define LDS_STRIDE 136       // 3 pad | halo col @3 | interior @4..131 | halo col @132
#define NPLANES (8 * 32)
#define PLANE_ELEMS (IMG_H * IMG_W)

typedef float v4f __attribute__((ext_vector_type(4)));

__device__ __forceinline__ uint32_t lds_off_u32(const void* p) {
  // Flat shared-memory addresses carry the LDS byte offset in the low 32 bits
  // (ISA 10.2); hardware adds the wave's LDS base.
  return (uint32_t)(uintptr_t)p;
}

__device__ __forceinline__ void async_load_b128(uint32_t lds_byte_off, const float* gptr) {
  asm volatile("global_load_async_to_lds_b128 %0, %1, off"
               :
               : "v"(lds_byte_off), "v"((uint64_t)(uintptr_t)gptr)
               : "memory");
}

__device__ __forceinline__ void wait_asynccnt0() {
#if __has_builtin(__builtin_amdgcn_s_wait_asynccnt)
  __builtin_amdgcn_s_wait_asynccnt(0);
#else
  asm volatile("s_wait_asynccnt 0" ::: "memory");
#endif
}

__device__ __forceinline__ int clampi(int v, int lo, int hi) {
  return v < lo ? lo : (v > hi ? hi : v);
}

__global__ __launch_bounds__(256) void erosion_and_dilation_74345883894463_kernel(
    const float* __restrict__ x, float* __restrict__ out) {
  __shared__ __align__(16) float tile[LDS_ROWS * LDS_STRIDE];

  const int tid = threadIdx.x;
  const int w0 = blockIdx.x * TILE_W;
  const int h0 = blockIdx.y * TILE_H;
  const int plane = blockIdx.z;
  const float* __restrict__ xp = x + (size_t)plane * PLANE_ELEMS;

  // ---- Halo columns (w0-1 and w0+TILE_W): 2 cols x 18 rows via clamped
  // global load -> ds_store (replicate border == inf-pad for min/max).
  if (tid < 2 * LDS_ROWS) {
    const int r = tid >> 1;
    const int side = tid & 1;
    const int gh = clampi(h0 - 1 + r, 0, IMG_H - 1);
    const int gw = clampi(side ? (w0 + TILE_W) : (w0 - 1), 0, IMG_W - 1);
    tile[r * LDS_STRIDE + (side ? (4 + TILE_W) : 3)] = xp[gh * IMG_W + gw];
  }

  // ---- Interior: 18 rows x 32 float4 = 576 async b128 copies (576 = 2*256 + 64,
  // trip-count split is wave-uniform: waves 0-1 issue 3, waves 2-7 issue 2).
  for (int i = tid; i < LDS_ROWS * (TILE_W / 4); i += 256) {
    const int r = i >> 5;        // 0..17
    const int c4 = i & 31;       // 0..31
    const int gh = clampi(h0 - 1 + r, 0, IMG_H - 1);
    const float* gp = xp + (size_t)gh * IMG_W + w0 + c4 * 4;             // 16B aligned
    const uint32_t lo = lds_off_u32(&tile[r * LDS_STRIDE + 4 + c4 * 4]); // 16B aligned
    async_load_b128(lo, gp);
  }

  wait_asynccnt0();
  __syncthreads();

  // ---- Compute: each thread produces two float4s of dilated and eroded outputs
  // (rows orow and orow+8 of the 16-row tile).
  const int orow0 = tid >> 5;    // 0..7
  const int cg = tid & 31;       // 0..31
  const int base = 4 + cg * 4;

  const size_t plane_off = (size_t)plane * PLANE_ELEMS;
  float* __restrict__ out_min = out + (size_t)NPLANES * PLANE_ELEMS;

#pragma unroll
  for (int half = 0; half < 2; ++half) {
    const int orow = orow0 + half * 8;   // 0..15

    v4f vmx, vmn;
#pragma unroll
    for (int r = 0; r < 3; ++r) {
      const float* rp = &tile[(orow + r) * LDS_STRIDE];
      const v4f v = *(const v4f*)(rp + base);  // aligned ds_load_b128
      const float l = rp[base - 1];
      const float q = rp[base + 4];

      v4f hx, hn;
      hx.x = fmaxf(fmaxf(l, v.x), v.y);
      hx.y = fmaxf(fmaxf(v.x, v.y), v.z);
      hx.z = fmaxf(fmaxf(v.y, v.z), v.w);
      hx.w = fmaxf(fmaxf(v.z, v.w), q);
      hn.x = fminf(fminf(l, v.x), v.y);
      hn.y = fminf(fminf(v.x, v.y), v.z);
      hn.z = fminf(fminf(v.y, v.z), v.w);
      hn.w = fminf(fminf(v.z, v.w), q);

      if (r == 0) {
        vmx = hx; vmn = hn;
      } else {
        vmx.x = fmaxf(vmx.x, hx.x); vmx.y = fmaxf(vmx.y, hx.y);
        vmx.z = fmaxf(vmx.z, hx.z); vmx.w = fmaxf(vmx.w, hx.w);
        vmn.x = fminf(vmn.x, hn.x); vmn.y = fminf(vmn.y, hn.y);
        vmn.z = fminf(vmn.z, hn.z); vmn.w = fminf(vmn.w, hn.w);
      }
    }

    const size_t obase = plane_off + (size_t)(h0 + orow) * IMG_W + (w0 + cg * 4);
    // Reference returns (dilated, eroded): dilated first, eroded at offset N.
    // Non-temporal: pure streaming writes, keep L2 for input-row reuse.
    __builtin_nontemporal_store(vmx, (v4f*)(out + obase));
    __builtin_nontemporal_store(vmn, (v4f*)(out_min + obase));
  }
}

extern "C" void kernel_launch(void* const* d_in, const int* in_sizes, int n_in,
                              void* d_out, int out_size, void* d_ws, size_t ws_size,
                              hipStream_t stream) {
  (void)in_sizes; (void)n_in; (void)out_size; (void)d_ws; (void)ws_size;
  const float* x = (const float*)d_in[0];
  float* out = (float*)d_out;

  dim3 grid(IMG_W / TILE_W, IMG_H / TILE_H, NPLANES);  // (4, 32, 256)
  dim3 block(256);
  erosion_and_dilation_74345883894463_kernel<<<grid, block, 0, stream>>>(x, out);
}